// Parametered_Graph_43860206027551
// MI455X (gfx1250) — compile-verified
//
#include <hip/hip_runtime.h>
#include <hip/hip_bf16.h>
#include <math.h>

// Problem geometry
#define NDIM  8192
#define TILE  128
#define NB    (NDIM / TILE)            // 64 tile rows/cols
#define NPAIR (NB * (NB + 1) / 2)      // 2080 lower-triangular tile pairs
#define LDP   130                      // padded LDS stride (floats): conflict-free
                                       // direct (2r+c) and transposed (2c+r) access

typedef float        v2f __attribute__((ext_vector_type(2)));
typedef float        v8f __attribute__((ext_vector_type(8)));
typedef unsigned int v4u __attribute__((ext_vector_type(4)));
typedef int          v4i __attribute__((ext_vector_type(4)));
typedef int          v8i __attribute__((ext_vector_type(8)));

#if __has_builtin(__builtin_amdgcn_tensor_load_to_lds) && \
    __has_builtin(__builtin_amdgcn_s_wait_tensorcnt)
#define USE_TDM 1
#else
#define USE_TDM 0
#endif

// Decode linear pair index -> (bi, bj) with bi >= bj
__device__ __forceinline__ void tri_decode(int idx, int& bi, int& bj) {
    float f = sqrtf(8.0f * (float)idx + 1.0f);
    int r = (int)((f - 1.0f) * 0.5f);
    while ((r + 1) * (r + 2) / 2 <= idx) ++r;
    while (r * (r + 1) / 2 > idx) --r;
    bi = r;
    bj = idx - r * (r + 1) / 2;
}

#if USE_TDM
// One-shot TDM DMA of a 128x128 f32 tile (row stride NDIM floats) from global
// memory into LDS with a 130-float padded row stride, per ISA 8.3/8.4:
//   group0: count=1 | lds_addr | global_addr[56:0] | type=2
//   group1: data_size=4B, pad_enable, pad_interval=128 DW, pad_amount=2 DW,
//           tensor_dim0/1=128, tile_dim0/1=128, tensor_dim0_stride=8192
//   groups 2/3: zero (2D tensor); trailing int32x8 operand of the 6-arg
//   builtin is zero-filled (matches the probe-verified call shape).
__device__ __forceinline__ void tdm_load_tile(const float* __restrict__ gsrc,
                                              float* lds_dst) {
    unsigned long long ga = (unsigned long long)gsrc;
    // Generic pointers to LDS carry the LDS byte offset in their low 32 bits.
    unsigned int la = (unsigned int)(unsigned long long)lds_dst;

    v4u g0;
    g0.x = 1u;                                          // count=1, is_restore=0
    g0.y = la;                                          // lds_addr (bytes)
    g0.z = (unsigned int)(ga & 0xFFFFFFFFu);            // global_addr[31:0]
    g0.w = (unsigned int)((ga >> 32) & 0x01FFFFFFu)     // global_addr[56:32]
         | (2u << 30);                                  // type = 2 ("image")

    v8i g1;
    g1[0] = (2 << 16)        // data_size = 2 -> 4 bytes
          | (1 << 20)        // pad_enable
          | (6 << 22)        // pad_interval code 6 -> every 128 DWORDs
          | (1 << 25);       // pad_amount  code 1 -> 2 DWORDs  (=> 130-float rows)
    g1[1] = (int)(128u << 16);   // tensor_dim0[15:0]=128 (in [31:16] of this dword)
    g1[2] = (int)(128u << 16);   // tensor_dim0 hi=0 | tensor_dim1[15:0]=128
    g1[3] = (int)(128u << 16);   // tensor_dim1 hi=0 | tile_dim0=128
    g1[4] = 128;                 // tile_dim1=128, tile_dim2=0
    g1[5] = NDIM;                // tensor_dim0_stride[31:0] = 8192
    g1[6] = 0;                   // stride hi | tensor_dim1_stride lo (unused, 2D)
    g1[7] = 0;

    v4i z4 = {0, 0, 0, 0};
    v8i z8 = {0, 0, 0, 0, 0, 0, 0, 0};
    __builtin_amdgcn_tensor_load_to_lds(g0, g1, z4, z4, z8, 0);
}
#endif

// Stage A(bj,bi) tile into LDS (row-major, LDP stride). Caller must
// __syncthreads() afterwards.
__device__ __forceinline__ void stage_tile(const float* __restrict__ src,
                                           float* lds, int tid) {
#if USE_TDM
    if (tid < 32) {                       // one wave issues the DMA; EXEC ignored
        tdm_load_tile(src, lds);
        __builtin_amdgcn_s_wait_tensorcnt((short)0);
    }
#else
    const int halft = tid >> 7;
    const int cc    = tid & 127;
    #pragma unroll 4
    for (int it = 0; it < TILE / 2; ++it) {
        int r = 2 * it + halft;
        lds[r * LDP + cc] = src[r * NDIM + cc];
    }
#endif
}

// -------------------------------------------------------------------------
// Pass 1: per-tile-pair partial row sums of adj = relu((A + A^T)/2), via
// V_WMMA_F32_16X16X4_F32 with an all-ones B operand (exact f32 reduction).
// partials[(bi*NB+bj)*TILE + m] = sum over column-tile bj of adj row (bi*TILE+m).
// Every one of the NB*NB*TILE slots is written exactly once -> deterministic.
// -------------------------------------------------------------------------
__global__ __launch_bounds__(256) void rowsum_pass(const float* __restrict__ A,
                                                   float* __restrict__ partials) {
    __shared__ float lt[TILE * LDP];   // A(bj,bi) tile, row-major
    __shared__ float sm[TILE * LDP];   // s = relu(0.5*(A_ij + A_ji))

    int bi, bj;
    tri_decode((int)blockIdx.x, bi, bj);

    const int tid   = (int)threadIdx.x;
    const int halft = tid >> 7;        // 0..1 (two rows per iteration)
    const int cc    = tid & 127;       // column within tile

    stage_tile(A + (bj * TILE) * NDIM + bi * TILE, lt, tid);
    __syncthreads();

    // Build s-tile: s[i][j] = relu(0.5*(A[bi+i][bj+j] + A[bj+j][bi+i])).
    // Transposed LDS read lt[cc*LDP + i] is bank-conflict-free (stride 130).
    {
        const float* src = A + (bi * TILE) * NDIM + bj * TILE;
        #pragma unroll 4
        for (int it = 0; it < TILE / 2; ++it) {
            int i = 2 * it + halft;
            float g = src[i * NDIM + cc];
            float t = lt[cc * LDP + i];
            float s = 0.5f * (g + t);
            sm[i * LDP + cc] = s > 0.0f ? s : 0.0f;
        }
    }
    __syncthreads();

    const int wave = tid >> 5;          // 8 waves, one 16-row strip each
    const int lane = tid & 31;
    const int mloc = lane & 15;         // M position in A fragment
    const int kp   = (lane >> 4) << 1;  // lanes 0-15 hold K={0,1}, 16-31 hold K={2,3}
    const v2f ones = {1.0f, 1.0f};      // B = ones(4x16): D[m][n] = sum_k A[m][k]

    // Row sums: contribute to global rows bi*TILE + [16w, 16w+16)
    {
        v8f acc = {0.f, 0.f, 0.f, 0.f, 0.f, 0.f, 0.f, 0.f};
        const int rowbase = (16 * wave + mloc) * LDP;
        #pragma unroll
        for (int kb = 0; kb < TILE; kb += 4) {
            v2f a;
            a.x = sm[rowbase + kb + kp];
            a.y = sm[rowbase + kb + kp + 1];
            acc = __builtin_amdgcn_wmma_f32_16x16x4_f32(
                false, a, false, ones, (short)0, acc, false, false);
        }
        // C/D layout: VGPR v, lanes 0-15 -> M=v (N=lane), lanes 16-31 -> M=v+8.
        float* dst = partials + (bi * NB + bj) * TILE + 16 * wave;
        if (lane == 0) {
            #pragma unroll
            for (int v = 0; v < 8; ++v) dst[v] = acc[v];
        } else if (lane == 16) {
            #pragma unroll
            for (int v = 0; v < 8; ++v) dst[8 + v] = acc[v];
        }
    }

    // Column sums (= row sums of s^T): contribute to rows bj*TILE + [16w,16w+16).
    // Uniform branch -> EXEC stays all-ones around the WMMAs.
    if (bi != bj) {
        v8f acc = {0.f, 0.f, 0.f, 0.f, 0.f, 0.f, 0.f, 0.f};
        const int colbase = 16 * wave + mloc;
        #pragma unroll
        for (int kb = 0; kb < TILE; kb += 4) {
            v2f a;
            a.x = sm[(kb + kp) * LDP + colbase];
            a.y = sm[(kb + kp + 1) * LDP + colbase];
            acc = __builtin_amdgcn_wmma_f32_16x16x4_f32(
                false, a, false, ones, (short)0, acc, false, false);
        }
        float* dst = partials + (bj * NB + bi) * TILE + 16 * wave;
        if (lane == 0) {
            #pragma unroll
            for (int v = 0; v < 8; ++v) dst[v] = acc[v];
        } else if (lane == 16) {
            #pragma unroll
            for (int v = 0; v < 8; ++v) dst[8 + v] = acc[v];
        }
    }
}

// -------------------------------------------------------------------------
// Pass 2: rowsum -> r_inv = rsqrt(rowsum + 1)  (self-loop; rowsum >= 0 so
// total >= 1 and the reference's inf-guard can never trigger, kept anyway).
// -------------------------------------------------------------------------
__global__ void rinv_pass(const float* __restrict__ partials,
                          float* __restrict__ rinv) {
    int i = (int)blockIdx.x * (int)blockDim.x + (int)threadIdx.x;
    if (i >= NDIM) return;
    const float* p = partials + ((i >> 7) * NB) * TILE + (i & 127);
    float s = 0.0f;
    #pragma unroll 8
    for (int t = 0; t < NB; ++t) s += p[t * TILE];
    float r = s + 1.0f;
    rinv[i] = (r > 0.0f) ? (1.0f / sqrtf(r)) : 0.0f;
}

// -------------------------------------------------------------------------
// Pass 3: out[i][j] = r_inv[i] * (relu((A_ij+A_ji)/2) + (i==j)) * r_inv[j].
// Output is symmetric: compute the (bi,bj) tile once, write it coalesced,
// and replay the mirror tile from LDS (conflict-free transposed reads).
// -------------------------------------------------------------------------
__global__ __launch_bounds__(256) void normalize_pass(const float* __restrict__ A,
                                                      const float* __restrict__ rinv,
                                                      float* __restrict__ out) {
    __shared__ float lt[TILE * LDP];   // A(bj,bi) tile
    __shared__ float st[TILE * LDP];   // finished output tile (bi,bj)

    int bi, bj;
    tri_decode((int)blockIdx.x, bi, bj);

    const int tid   = (int)threadIdx.x;
    const int halft = tid >> 7;
    const int cc    = tid & 127;

    stage_tile(A + (bj * TILE) * NDIM + bi * TILE, lt, tid);
    __syncthreads();

    {
        const float* src  = A + (bi * TILE) * NDIM + bj * TILE;
        float*       dstg = out + (bi * TILE) * NDIM + bj * TILE;
        const int    gj   = bj * TILE + cc;
        const float  rj   = rinv[gj];
        #pragma unroll 2
        for (int it = 0; it < TILE / 2; ++it) {
            int i  = 2 * it + halft;
            int gi = bi * TILE + i;
            if (it + 1 < TILE / 2)
                __builtin_prefetch(src + (i + 2) * NDIM + cc, 0, 0); // global_prefetch_b8
            float g = src[i * NDIM + cc];
            float t = lt[cc * LDP + i];
            float s = 0.5f * (g + t);
            s = s > 0.0f ? s : 0.0f;
            if (gi == gj) s += 1.0f;              // + I (only on diagonal tiles)
            float val = rinv[gi] * rj * s;
            dstg[i * NDIM + cc] = val;            // coalesced write of tile (bi,bj)
            st[i * LDP + cc]    = val;
        }
    }

    if (bi != bj) {
        __syncthreads();
        float* dstg = out + (bj * TILE) * NDIM + bi * TILE;
        #pragma unroll 4
        for (int it = 0; it < TILE / 2; ++it) {
            int r2 = 2 * it + halft;
            dstg[r2 * NDIM + cc] = st[cc * LDP + r2]; // mirror tile, coalesced write
        }
    }
}

// -------------------------------------------------------------------------
extern "C" void kernel_launch(void* const* d_in, const int* in_sizes, int n_in,
                              void* d_out, int out_size, void* d_ws, size_t ws_size,
                              hipStream_t stream) {
    (void)in_sizes; (void)n_in; (void)out_size; (void)ws_size;
    const float* A   = (const float*)d_in[0];
    float*       out = (float*)d_out;

    // Workspace: 64*64*128 partial row sums (2 MB) + 8192 r_inv values.
    float* partials = (float*)d_ws;
    float* rinv     = partials + NB * NB * TILE;

    rowsum_pass   <<<NPAIR,      256, 0, stream>>>(A, partials);
    rinv_pass     <<<NDIM / 256, 256, 0, stream>>>(partials, rinv);
    normalize_pass<<<NPAIR,      256, 0, stream>>>(A, rinv, out);
}